// YOLOHead_14001593385147
// MI455X (gfx1250) — compile-verified
//
#include <hip/hip_runtime.h>
#include <math.h>

typedef float v2f __attribute__((ext_vector_type(2)));
typedef float v8f __attribute__((ext_vector_type(8)));

#define NUM_CLASSES 5
#define NA 3
#define N_TOTAL 25200   // 3*6400 + 3*1600 + 3*400
#define MAX_DET 300
#define CONF_THR 0.25f
#define IOU_THR 0.45f
#define NBATCH 16

__device__ __forceinline__ float sigm(float x) { return 1.0f / (1.0f + expf(-x)); }

// ---------------------------------------------------------------------------
// Fused 1x1-conv (WMMA f32 16x16x4 GEMM) + YOLO decode.
// One wave -> 16 spatial positions x 32 (padded) output channels.
// 8 waves / block. Weights staged in LDS (padded to 32 rows).
// ---------------------------------------------------------------------------
__global__ __launch_bounds__(256) void head_kernel(
    const float* __restrict__ x,     // (B, C, H, W)
    const float* __restrict__ w,     // (30, C)
    const float* __restrict__ bias,  // (30,)
    float* __restrict__ dets,        // (B, N_TOTAL, 6)
    int C, int H, int W, float stride_, int det_off,
    float a0w, float a0h, float a1w, float a1h, float a2w, float a2h)
{
    const int HW = H * W;
    const int tiles_per_b = HW >> 4;          // HW/16, exact for all heads
    const int wave = threadIdx.x >> 5;
    const int lane = threadIdx.x & 31;
    const int tile_id = blockIdx.x * 8 + wave;
    const int b  = tile_id / tiles_per_b;
    const int t  = tile_id - b * tiles_per_b;
    const int s0 = t << 4;                     // first spatial index of tile

    extern __shared__ float sm[];
    float* wpad  = sm;                 // [32][C], rows 30,31 zero
    float* bpad  = sm + 32 * C;        // [32]
    float* tiles = bpad + 32;          // [8][32][16]

    // cooperative weight/bias staging
    for (int idx = threadIdx.x; idx < 32 * C; idx += 256) {
        int m = idx / C;
        int c = idx - m * C;
        wpad[idx] = (m < 30) ? w[m * C + c] : 0.0f;
    }
    if (threadIdx.x < 32)
        bpad[threadIdx.x] = (threadIdx.x < 30) ? bias[threadIdx.x] : 0.0f;
    __syncthreads();

    // --- WMMA GEMM: D(16x16) += A(16x4) * B(4x16), K stepped by 4 ---
    const int half = lane >> 4;        // 0: lanes 0-15 (K pair 0,1) | 1: lanes 16-31 (K pair 2,3)
    const int lr   = lane & 15;        // row (A) / col (B) index
    const float* xb = x + (size_t)b * C * HW;
    const int sp0 = s0 + lr;

    v8f acc0 = {0.f,0.f,0.f,0.f,0.f,0.f,0.f,0.f};
    v8f acc1 = {0.f,0.f,0.f,0.f,0.f,0.f,0.f,0.f};

    for (int k0 = 0; k0 < C; k0 += 4) {
        int c0 = k0 + 2 * half;
        v2f bvec;
        bvec.x = xb[(size_t)c0 * HW + sp0];
        bvec.y = xb[(size_t)(c0 + 1) * HW + sp0];
        v2f av0 = *(const v2f*)(wpad + lr * C + c0);          // oc tile 0 (rows 0-15)
        v2f av1 = *(const v2f*)(wpad + (16 + lr) * C + c0);   // oc tile 1 (rows 16-31)
        acc0 = __builtin_amdgcn_wmma_f32_16x16x4_f32(false, av0, false, bvec,
                                                     (short)0, acc0, false, false);
        acc1 = __builtin_amdgcn_wmma_f32_16x16x4_f32(false, av1, false, bvec,
                                                     (short)0, acc1, false, false);
    }

    // spill D tiles to LDS: [ch][pos], ch = out-channel (0..31), pos = 0..15
    float* my_tile = tiles + wave * 512;
    #pragma unroll
    for (int r = 0; r < 8; ++r) {
        my_tile[(r + 8 * half) * 16 + lr]        = acc0[r];
        my_tile[(16 + r + 8 * half) * 16 + lr]   = acc1[r];
    }
    __syncthreads();

    // --- decode: 48 anchor-positions per tile ---
    float aw[3] = {a0w, a1w, a2w};
    float ah[3] = {a0h, a1h, a2h};

    for (int ap = lane; ap < 48; ap += 32) {
        int a = ap >> 4;
        int s = ap & 15;
        int sp = s0 + s;
        int gy = sp / W;
        int gx = sp - gy * W;

        float p[10];
        #pragma unroll
        for (int c = 0; c < 10; ++c)
            p[c] = my_tile[(a * 10 + c) * 16 + s] + bpad[a * 10 + c];

        float xx = (sigm(p[0]) + (float)gx) * stride_;
        float yy = (sigm(p[1]) + (float)gy) * stride_;
        float wv = expf(p[2]) * aw[a];
        float hv = expf(p[3]) * ah[a];
        float obj = sigm(p[4]);
        float m = -1.0f; int id = 0;
        #pragma unroll
        for (int c = 0; c < NUM_CLASSES; ++c) {
            float sc = sigm(p[5 + c]);
            if (sc > m) { m = sc; id = c; }   // strict > keeps first max (argmax semantics)
        }
        float conf = obj * m;

        size_t n = (size_t)b * N_TOTAL + det_off + (size_t)a * HW + sp;
        float* o = dets + n * 6;
        o[0] = xx; o[1] = yy; o[2] = wv; o[3] = hv; o[4] = conf; o[5] = (float)id;
    }
}

// ---------------------------------------------------------------------------
// NMS: one 1024-thread workgroup per batch image.
// Scores live in LDS (25200 f32 ~ 101 KB, fits 320 KB WGP LDS).
// Matches reference: 300 fixed iterations, argmax with first-index
// tie-break, suppression gated on validity, kept = dets[j]*valid.
// ---------------------------------------------------------------------------
__global__ __launch_bounds__(1024) void nms_kernel(
    const float* __restrict__ dets,  // (B, N_TOTAL, 6)
    float* __restrict__ kept,        // (B, 300, 6)
    float* __restrict__ valid)       // (B, 300)
{
    extern __shared__ float sm[];
    float* scores = sm;                       // [N_TOTAL]
    float* rv = sm + N_TOTAL;                 // [1024]
    int*   ri = (int*)(sm + N_TOTAL + 1024);  // [1024]

    const int b = blockIdx.x;
    const int tid = threadIdx.x;
    const float* db = dets + (size_t)b * N_TOTAL * 6;

    for (int i = tid; i < N_TOTAL; i += 1024) {
        float c = db[(size_t)i * 6 + 4];
        scores[i] = (c > CONF_THR) ? c : -1.0f;
    }
    __syncthreads();

    for (int it = 0; it < MAX_DET; ++it) {
        // parallel argmax (first max wins)
        float best = -3.4e38f; int bi = 0x7fffffff;
        for (int i = tid; i < N_TOTAL; i += 1024) {
            float v = scores[i];
            if (v > best) { best = v; bi = i; }
        }
        rv[tid] = best; ri[tid] = bi;
        __syncthreads();
        for (int s2 = 512; s2 > 0; s2 >>= 1) {
            if (tid < s2) {
                float v2 = rv[tid + s2]; int i2 = ri[tid + s2];
                if (v2 > rv[tid] || (v2 == rv[tid] && i2 < ri[tid])) {
                    rv[tid] = v2; ri[tid] = i2;
                }
            }
            __syncthreads();
        }
        int j = ri[0];
        int v = rv[0] > 0.0f;

        // box j (redundant per-thread; hits L2)
        float xj = db[(size_t)j * 6 + 0], yj = db[(size_t)j * 6 + 1];
        float wj = db[(size_t)j * 6 + 2], hj = db[(size_t)j * 6 + 3];
        float x1j = xj - wj * 0.5f, y1j = yj - hj * 0.5f;
        float x2j = x1j + wj,       y2j = y1j + hj;
        float aj  = wj * hj;

        if (tid < 6)
            kept[((size_t)b * MAX_DET + it) * 6 + tid] = v ? db[(size_t)j * 6 + tid] : 0.0f;
        if (tid == 0)
            valid[(size_t)b * MAX_DET + it] = v ? 1.0f : 0.0f;

        if (v) {
            for (int i = tid; i < N_TOTAL; i += 1024) {
                float sv = scores[i];
                if (sv > -1.0f) {
                    float xi = db[(size_t)i * 6 + 0], yi = db[(size_t)i * 6 + 1];
                    float wi = db[(size_t)i * 6 + 2], hi = db[(size_t)i * 6 + 3];
                    float x1 = xi - wi * 0.5f, y1 = yi - hi * 0.5f;
                    float x2 = x1 + wi,        y2 = y1 + hi;
                    float ix1 = fmaxf(x1j, x1), iy1 = fmaxf(y1j, y1);
                    float ix2 = fminf(x2j, x2), iy2 = fminf(y2j, y2);
                    float inter = fmaxf(ix2 - ix1, 0.0f) * fmaxf(iy2 - iy1, 0.0f);
                    float iou = inter / (aj + wi * hi - inter + 1e-7f);
                    if (iou >= IOU_THR) scores[i] = -1.0f;
                }
            }
        }
        __syncthreads();
    }
}

extern "C" void kernel_launch(void* const* d_in, const int* in_sizes, int n_in,
                              void* d_out, int out_size, void* d_ws, size_t ws_size,
                              hipStream_t stream) {
    (void)in_sizes; (void)n_in; (void)out_size; (void)d_ws; (void)ws_size;

    const float* p3 = (const float*)d_in[0];
    const float* p4 = (const float*)d_in[1];
    const float* p5 = (const float*)d_in[2];
    const float* w3 = (const float*)d_in[3];
    const float* b3 = (const float*)d_in[4];
    const float* w4 = (const float*)d_in[5];
    const float* b4 = (const float*)d_in[6];
    const float* w5 = (const float*)d_in[7];
    const float* b5 = (const float*)d_in[8];

    float* dets  = (float*)d_out;                             // 16*25200*6
    float* kept  = dets + (size_t)NBATCH * N_TOTAL * 6;       // 16*300*6
    float* valid = kept + (size_t)NBATCH * MAX_DET * 6;       // 16*300

    // shared mem: 32*C weights + 32 bias + 8 waves * 32*16 tile
    size_t sh3 = (size_t)(32 * 96  + 32 + 8 * 512) * sizeof(float);
    size_t sh4 = (size_t)(32 * 192 + 32 + 8 * 512) * sizeof(float);
    size_t sh5 = (size_t)(32 * 384 + 32 + 8 * 512) * sizeof(float);

    // tiles = B * HW/16 ; blocks = tiles/8 (exact)
    head_kernel<<<800, 256, sh3, stream>>>(p3, w3, b3, dets, 96, 80, 80, 8.0f, 0,
                                           10.f, 13.f, 16.f, 30.f, 33.f, 23.f);
    head_kernel<<<200, 256, sh4, stream>>>(p4, w4, b4, dets, 192, 40, 40, 16.0f, 19200,
                                           30.f, 61.f, 62.f, 45.f, 59.f, 119.f);
    head_kernel<<<50,  256, sh5, stream>>>(p5, w5, b5, dets, 384, 20, 20, 32.0f, 24000,
                                           116.f, 90.f, 156.f, 198.f, 373.f, 326.f);

    size_t shn = (size_t)(N_TOTAL + 1024 + 1024) * sizeof(float);
    nms_kernel<<<NBATCH, 1024, shn, stream>>>(dets, kept, valid);
}